// NGNN_GCNConv_74904229642493
// MI455X (gfx1250) — compile-verified
//
#include <hip/hip_runtime.h>
#include <stdint.h>

// ---------------------------------------------------------------------------
// NGNN GCNConv + 2-layer MLP head, fp32 end-to-end with V_WMMA_F32_16X16X4_F32.
// Weights are pre-swizzled in LDS into K-pair layout so every WMMA B-fragment
// is one aligned ds_load_b64 (no VGPR repacking).  Weight staging uses the
// gfx1250 async global->LDS copy path (GLOBAL_LOAD_ASYNC_TO_LDS_B128).
// ---------------------------------------------------------------------------

typedef __attribute__((ext_vector_type(2))) float v2f;
typedef __attribute__((ext_vector_type(8))) float v8f;
typedef int v4i __attribute__((vector_size(4 * sizeof(int))));  // builtin's pointee type

// ---------------- async global->LDS copy (guarded) ----------------------
#if defined(__has_builtin)
#  if __has_builtin(__builtin_amdgcn_global_load_async_to_lds_b128)
#    define HAVE_ASYNC_LDS 1
#  endif
#endif
#ifndef HAVE_ASYNC_LDS
#  define HAVE_ASYNC_LDS 0
#endif

#define NN   100000   // nodes
#define NE   1000000  // edges
#define DIN  128
#define DH   64
#define DOUT 64
#define NTILES (NN / 16)   // 6250 16-row tiles, divides exactly

__device__ __forceinline__ void async_copy16(const float* g, float* l) {
#if HAVE_ASYNC_LDS
    // generic LDS addr low 32 bits == LDS offset (aperture rule); integer casts
    // are always-legal conversions into the address-space-qualified pointers.
    __builtin_amdgcn_global_load_async_to_lds_b128(
        (__attribute__((address_space(1))) v4i*)(uintptr_t)g,
        (__attribute__((address_space(3))) v4i*)(uint32_t)(uintptr_t)l,
        0, 0);
#else
    *(float4*)l = *(const float4*)g;
#endif
}

__device__ __forceinline__ void async_join() {
#if HAVE_ASYNC_LDS
    asm volatile("s_wait_asynccnt 0x0" ::: "memory");
#endif
}

// Stage W (rows x 64, row-major) into pair layout: sB[p*64+c] = {W[2p][c], W[2p+1][c]}.
// Chunked through an 8KB sRaw staging buffer.  blockDim.x == 128.
__device__ __forceinline__ void load_pairs(const float* __restrict__ W, v2f* sB,
                                           float* sRaw, int t, int rows) {
    for (int c4 = 0; c4 < (rows >> 5); ++c4) {          // 32 rows / chunk
        for (int i = t * 4; i < 32 * 64; i += 128 * 4)  // 8KB async copy
            async_copy16(W + c4 * 2048 + i, sRaw + i);
        async_join();
        __syncthreads();
        for (int i = t; i < 16 * 64; i += 128) {        // build 16 pair-rows
            int pl = i >> 6, c = i & 63;
            v2f tt;
            tt[0] = sRaw[(2 * pl) * 64 + c];
            tt[1] = sRaw[(2 * pl + 1) * 64 + c];
            sB[(c4 * 16 + pl) * 64 + c] = tt;
        }
        __syncthreads();
    }
}

// ---------------- degree / normalization --------------------------------
__global__ void k_init_deg(float* deg) {
    int i = blockIdx.x * blockDim.x + threadIdx.x;
    if (i < NN) deg[i] = 1.0f;                       // self-loop
}

__global__ void k_deg_accum(const int* __restrict__ dst, float* deg) {
    int e = blockIdx.x * blockDim.x + threadIdx.x;
    if (e < NE) atomicAdd(&deg[dst[e]], 1.0f);
}

__global__ void k_dinv(float* deg) {
    int i = blockIdx.x * blockDim.x + threadIdx.x;
    if (i < NN) deg[i] = rsqrtf(deg[i]);             // deg >= 1 always
}

// ---------------- GEMM1: h = x @ W_conv  ([NN,128] x [128,64]) ----------
// One wave -> 16x64 tile, 4 f32 WMMA accumulators, K stepped by 4.
// A-frag: lane l<16 -> row l, K pair (kk,kk+1); lane>=16 -> row l-16, (kk+2,kk+3).
// B-frag: single ds_load_b64 from pair-layout sB.
__global__ __launch_bounds__(128) void k_gemm1(const float* __restrict__ x,
                                               const float* __restrict__ W,
                                               float* __restrict__ h) {
    __shared__ v2f   sB[(DIN / 2) * DH];             // 32 KB pair layout
    __shared__ float sRaw[32 * DH];                  // 8 KB staging
    const int t = threadIdx.x;
    load_pairs(W, sB, sRaw, t, DIN);

    const int wave  = t >> 5;
    const int lane  = t & 31;
    const int tile  = blockIdx.x * 4 + wave;
    if (tile >= NTILES) return;                      // wave-uniform
    const int row0  = tile * 16;
    const int lcol  = lane & 15;
    const int halfK = (lane >> 4) * 2;               // 0 or 2
    const int pofs  = lane >> 4;                     // pair-row offset

    v8f a0 = {}, a1 = {}, a2 = {}, a3 = {};
    const float* xrow = x + (size_t)(row0 + lcol) * DIN;
    for (int kk = 0; kk < DIN; kk += 4) {
        v2f a = *(const v2f*)(xrow + kk + halfK);    // 8B-aligned global b64
        const v2f* bp = sB + ((kk >> 1) + pofs) * DH + lcol;
        v2f b0 = bp[0], b1 = bp[16], b2 = bp[32], b3 = bp[48];
        a0 = __builtin_amdgcn_wmma_f32_16x16x4_f32(false, a, false, b0, (short)0, a0, false, false);
        a1 = __builtin_amdgcn_wmma_f32_16x16x4_f32(false, a, false, b1, (short)0, a1, false, false);
        a2 = __builtin_amdgcn_wmma_f32_16x16x4_f32(false, a, false, b2, (short)0, a2, false, false);
        a3 = __builtin_amdgcn_wmma_f32_16x16x4_f32(false, a, false, b3, (short)0, a3, false, false);
    }
    // C layout: vgpr v, lane l -> row = v + 8*(l>=16), col = (l&15) + 16*nt
    const int rb = row0 + 8 * (lane >> 4);
    for (int v = 0; v < 8; ++v) {
        float* op = h + (size_t)(rb + v) * DH + lcol;
        op[0] = a0[v]; op[16] = a1[v]; op[32] = a2[v]; op[48] = a3[v];
    }
}

// ---------------- agg = b_conv + h * dinv^2  (self-loop term) -----------
__global__ void k_agg_init(const float* __restrict__ h, const float* __restrict__ dinv,
                           const float* __restrict__ bc, float* __restrict__ agg) {
    int tid = blockIdx.x * blockDim.x + threadIdx.x;         // NN*16
    if (tid >= NN * 16) return;
    int node = tid >> 4;
    int f4   = (tid & 15) * 4;
    float di = dinv[node];
    float s  = di * di;
    float4 hv = *(const float4*)(h + (size_t)node * DH + f4);
    float4 o;
    o.x = hv.x * s + bc[f4 + 0];
    o.y = hv.y * s + bc[f4 + 1];
    o.z = hv.z * s + bc[f4 + 2];
    o.w = hv.w * s + bc[f4 + 3];
    *(float4*)(agg + (size_t)node * DH + f4) = o;
}

// ---------------- edge scatter: agg[dst] += h[src] * dinv[s]*dinv[d] ----
__global__ void k_scatter(const float* __restrict__ h, const int* __restrict__ src,
                          const int* __restrict__ dst, const float* __restrict__ dinv,
                          float* __restrict__ agg) {
    int tid = blockIdx.x * blockDim.x + threadIdx.x;         // NE*16
    if (tid >= NE * 16) return;
    int e  = tid >> 4;
    int f4 = (tid & 15) * 4;
    int s = src[e], d = dst[e];
    float nrm = dinv[s] * dinv[d];
    float4 hv = *(const float4*)(h + (size_t)s * DH + f4);
    float* ap = agg + (size_t)d * DH + f4;
    atomicAdd(ap + 0, hv.x * nrm);
    atomicAdd(ap + 1, hv.y * nrm);
    atomicAdd(ap + 2, hv.z * nrm);
    atomicAdd(ap + 3, hv.w * nrm);
}

// ---------------- fused MLP: out = relu(relu(agg)@W1+b1)@W2+b2 ----------
__global__ __launch_bounds__(128) void k_mlp(const float* __restrict__ agg,
                                             const float* __restrict__ W1, const float* __restrict__ bb1,
                                             const float* __restrict__ W2, const float* __restrict__ bb2,
                                             float* __restrict__ out) {
    __shared__ v2f   sB1[(DH / 2) * DH];             // 16 KB pair layout
    __shared__ v2f   sB2[(DH / 2) * DOUT];           // 16 KB pair layout
    __shared__ float sRaw[32 * DH];                  // 8 KB staging (reused)
    __shared__ float sT[4][16 * DH];                 // 16 KB C->A restage
    const int t = threadIdx.x;
    load_pairs(W1, sB1, sRaw, t, DH);
    load_pairs(W2, sB2, sRaw, t, DH);

    const int wave  = t >> 5;
    const int lane  = t & 31;
    const int tile  = blockIdx.x * 4 + wave;
    const bool active = tile < NTILES;               // wave-uniform
    const int row0  = tile * 16;
    const int lcol  = lane & 15;
    const int halfK = (lane >> 4) * 2;
    const int pofs  = lane >> 4;
    const int mb    = 8 * (lane >> 4);

    if (active) {
        v8f f0 = {}, f1 = {}, f2 = {}, f3 = {};
        const float* arow = agg + (size_t)(row0 + lcol) * DH;
        for (int kk = 0; kk < DH; kk += 4) {
            v2f a;
            a[0] = fmaxf(arow[kk + halfK],     0.0f);   // relu(layer-1 input)
            a[1] = fmaxf(arow[kk + halfK + 1], 0.0f);
            const v2f* bp = sB1 + ((kk >> 1) + pofs) * DH + lcol;
            v2f b0 = bp[0], b1 = bp[16], b2 = bp[32], b3 = bp[48];
            f0 = __builtin_amdgcn_wmma_f32_16x16x4_f32(false, a, false, b0, (short)0, f0, false, false);
            f1 = __builtin_amdgcn_wmma_f32_16x16x4_f32(false, a, false, b1, (short)0, f1, false, false);
            f2 = __builtin_amdgcn_wmma_f32_16x16x4_f32(false, a, false, b2, (short)0, f2, false, false);
            f3 = __builtin_amdgcn_wmma_f32_16x16x4_f32(false, a, false, b3, (short)0, f3, false, false);
        }
        float c0 = bb1[lcol], c1 = bb1[16 + lcol], c2 = bb1[32 + lcol], c3 = bb1[48 + lcol];
        for (int v = 0; v < 8; ++v) {                // bias + relu, restage row-major
            float* sp = sT[wave] + (mb + v) * DH + lcol;
            sp[0]  = fmaxf(f0[v] + c0, 0.0f);
            sp[16] = fmaxf(f1[v] + c1, 0.0f);
            sp[32] = fmaxf(f2[v] + c2, 0.0f);
            sp[48] = fmaxf(f3[v] + c3, 0.0f);
        }
    }
    __syncthreads();
    if (active) {
        v8f g0 = {}, g1 = {}, g2 = {}, g3 = {};
        const float* trow = sT[wave] + lcol * DH;
        for (int kk = 0; kk < DH; kk += 4) {
            v2f a = *(const v2f*)(trow + kk + halfK);   // aligned ds_load_b64
            const v2f* bp = sB2 + ((kk >> 1) + pofs) * DOUT + lcol;
            v2f b0 = bp[0], b1 = bp[16], b2 = bp[32], b3 = bp[48];
            g0 = __builtin_amdgcn_wmma_f32_16x16x4_f32(false, a, false, b0, (short)0, g0, false, false);
            g1 = __builtin_amdgcn_wmma_f32_16x16x4_f32(false, a, false, b1, (short)0, g1, false, false);
            g2 = __builtin_amdgcn_wmma_f32_16x16x4_f32(false, a, false, b2, (short)0, g2, false, false);
            g3 = __builtin_amdgcn_wmma_f32_16x16x4_f32(false, a, false, b3, (short)0, g3, false, false);
        }
        float c0 = bb2[lcol], c1 = bb2[16 + lcol], c2 = bb2[32 + lcol], c3 = bb2[48 + lcol];
        const int rb = row0 + mb;
        for (int v = 0; v < 8; ++v) {
            float* op = out + (size_t)(rb + v) * DOUT + lcol;
            op[0]  = g0[v] + c0;
            op[16] = g1[v] + c1;
            op[32] = g2[v] + c2;
            op[48] = g3[v] + c3;
        }
    }
}

// ---------------------------------------------------------------------------
extern "C" void kernel_launch(void* const* d_in, const int* in_sizes, int n_in,
                              void* d_out, int out_size, void* d_ws, size_t ws_size,
                              hipStream_t stream) {
    const float* x   = (const float*)d_in[0];
    const int*   ei  = (const int*)d_in[1];      // (2, NE) flat
    const int*   src = ei;
    const int*   dst = ei + NE;
    const float* Wc  = (const float*)d_in[2];
    const float* bc  = (const float*)d_in[3];
    const float* W1  = (const float*)d_in[4];
    const float* b1  = (const float*)d_in[5];
    const float* W2  = (const float*)d_in[6];
    const float* b2  = (const float*)d_in[7];
    float*       out = (float*)d_out;
    (void)in_sizes; (void)n_in; (void)out_size; (void)ws_size;

    // scratch layout
    char*  ws  = (char*)d_ws;
    size_t off = 0;
    auto carve = [&](size_t bytes) -> void* {
        void* p = ws + off;
        off = (off + bytes + 255) & ~(size_t)255;
        return p;
    };
    float* deg = (float*)carve(sizeof(float) * NN);        // becomes dinv in-place
    float* h   = (float*)carve(sizeof(float) * NN * DH);
    float* agg = (float*)carve(sizeof(float) * NN * DH);

    k_init_deg <<<(NN + 255) / 256, 256, 0, stream>>>(deg);
    k_deg_accum<<<(NE + 255) / 256, 256, 0, stream>>>(dst, deg);
    k_dinv     <<<(NN + 255) / 256, 256, 0, stream>>>(deg);

    k_gemm1    <<<(NTILES + 3) / 4, 128, 0, stream>>>(x, Wc, h);

    k_agg_init <<<(NN * 16 + 255) / 256, 256, 0, stream>>>(h, deg, bc, agg);
    k_scatter  <<<(NE * 16 + 255) / 256, 256, 0, stream>>>(h, src, dst, deg, agg);

    k_mlp      <<<(NTILES + 3) / 4, 128, 0, stream>>>(agg, W1, b1, W2, b2, out);
}